// MMD_loss_46660524704258
// MI455X (gfx1250) — compile-verified
//
#include <hip/hip_runtime.h>

typedef __attribute__((ext_vector_type(2))) float v2f;
typedef __attribute__((ext_vector_type(8))) float v8f;

#define NX      4096
#define N_TOT   8192
#define DIM     64
#define MIN_BW  1e-12f
// 10 kernels (all bandwidths identical since KERNEL_MUL==1), 1/(4096*4096)
#define W_POS   (10.0f / 16777216.0f)

// ---------------- helpers ----------------
__device__ __forceinline__ const float* row_ptr(const float* __restrict__ src,
                                                const float* __restrict__ tgt,
                                                int i) {
    return (i < NX) ? (src + (size_t)i * DIM) : (tgt + (size_t)(i - NX) * DIM);
}

// ---------------- kernel 1: row norms ----------------
// grid 1024 x 256 threads: one wave32 per row of the concatenated 8192x64 matrix
__global__ void mmd_row_norms(const float* __restrict__ src,
                              const float* __restrict__ tgt,
                              float* __restrict__ norms) {
    int wave = threadIdx.x >> 5;
    int lane = threadIdx.x & 31;
    int row  = blockIdx.x * 8 + wave;
    const float* p = row_ptr(src, tgt, row);
    v2f a = *(const v2f*)(p + lane * 2);
    float s = a.x * a.x + a.y * a.y;
    #pragma unroll
    for (int off = 16; off; off >>= 1) s += __shfl_xor(s, off, 32);
    if (lane == 0) norms[row] = s;
}

// ---------------- kernel 2: bandwidth (single block, 256 threads) ----------------
// sum(d2_tot) = 2*n*sum(|t_i|^2) - 2*|sum_i t_i|^2   (diagonal contributes 0)
__global__ void mmd_bandwidth(const float* __restrict__ src,
                              const float* __restrict__ tgt,
                              const float* __restrict__ norms,
                              float* __restrict__ bw_out) {
    __shared__ float sm[256];
    int tid = threadIdx.x;

    // S2 = sum of row norms (fixed-order per thread + fixed tree)
    float s = 0.0f;
    for (int j = 0; j < 32; ++j) s += norms[tid + j * 256];
    sm[tid] = s;
    __syncthreads();
    for (int off = 128; off; off >>= 1) {
        if (tid < off) sm[tid] += sm[tid + off];
        __syncthreads();
    }
    float S2 = sm[0];
    __syncthreads();

    // column sums: k = column (0..63), c = row chunk (0..3, 2048 rows each)
    int k = tid & 63;
    int c = tid >> 6;
    const float* base = (c < 2) ? (src + (size_t)c * 2048 * DIM)
                                : (tgt + (size_t)(c - 2) * 2048 * DIM);
    float cs = 0.0f;
    for (int r = 0; r < 2048; ++r) cs += base[(size_t)r * DIM + k];
    sm[tid] = cs;
    __syncthreads();
    if (tid < 64) {
        float sk = sm[k] + sm[k + 64] + sm[k + 128] + sm[k + 192];
        sm[k] = sk * sk;
    }
    __syncthreads();
    if (tid == 0) {
        float s2n = 0.0f;
        for (int q = 0; q < 64; ++q) s2n += sm[q];
        float n = (float)N_TOT;
        float total_d2 = 2.0f * n * S2 - 2.0f * s2n;
        float bw = total_d2 / (n * n - n);
        bw_out[0] = fmaxf(bw, MIN_BW);
    }
}

// ---------------- kernel 3: main WMMA + exp kernel ----------------
// 512x512 tile grid of 16x16 tiles. 4096 blocks x 8 waves; each wave does 8 tiles.
// Per tile: 16 x V_WMMA_F32_16X16X4_F32 over K=64, then d2 + exp + weighted sum.
__global__ void mmd_main(const float* __restrict__ src,
                         const float* __restrict__ tgt,
                         const float* __restrict__ norms,
                         const float* __restrict__ bw_in,
                         float* __restrict__ partials) {
    int wave = threadIdx.x >> 5;
    int lane = threadIdx.x & 31;
    int gw   = blockIdx.x * 8 + wave;          // global wave id, 0..32767

    float bw     = bw_in[0];
    float negInv = -1.0f / bw;

    int laneM = lane & 15;
    int koff  = (lane < 16) ? 0 : 2;           // f32 A/B layout: VGPR0 K in {0,2}
    int mhalf = (lane < 16) ? 0 : 8;           // C/D layout: M = v + (lane<16?0:8)

    float waveAcc = 0.0f;

    for (int t = 0; t < 8; ++t) {
        int tile = gw + (t << 15);             // + t*32768
        int ti = tile >> 9;                    // row tile (0..511)
        int tj = tile & 511;                   // col tile
        int i0 = ti << 4;
        int j0 = tj << 4;

        const float* pa = row_ptr(src, tgt, i0 + laneM);  // A row (M = laneM)
        const float* pb = row_ptr(src, tgt, j0 + laneM);  // B col (N = laneM), B[k][n]=T[j0+n][k]

        v8f acc = {};
        #pragma unroll
        for (int kk = 0; kk < 16; ++kk) {
            v2f a = *(const v2f*)(pa + 4 * kk + koff);    // K = 4kk+koff, 4kk+koff+1
            v2f b = *(const v2f*)(pb + 4 * kk + koff);
            acc = __builtin_amdgcn_wmma_f32_16x16x4_f32(
                /*neg_a=*/false, a, /*neg_b=*/false, b,
                /*c_mod=*/(short)0, acc, /*reuse_a=*/false, /*reuse_b=*/false);
        }

        // d2 = |t_i|^2 + |t_j|^2 - 2 G, clamp, exp
        int ibase = i0 + mhalf;                            // 8 consecutive row norms
        const float4* np = (const float4*)(norms + ibase); // 32B aligned
        float4 n0 = np[0], n1 = np[1];
        float nrow[8] = { n0.x, n0.y, n0.z, n0.w, n1.x, n1.y, n1.z, n1.w };
        float nj = norms[j0 + laneM];

        float tileSum = 0.0f;
        #pragma unroll
        for (int v = 0; v < 8; ++v) {
            float d2 = nrow[v] + nj - 2.0f * acc[v];
            d2 = fmaxf(d2, 0.0f);
            tileSum += __expf(d2 * negInv);
        }

        bool isrc = (i0 < NX), jsrc = (j0 < NX);
        float w = (isrc == jsrc) ? W_POS : -W_POS;
        waveAcc += tileSum * w;
    }

    // deterministic intra-wave + intra-block reduction
    #pragma unroll
    for (int off = 16; off; off >>= 1) waveAcc += __shfl_xor(waveAcc, off, 32);

    __shared__ float smw[8];
    if (lane == 0) smw[wave] = waveAcc;
    __syncthreads();
    if (threadIdx.x == 0) {
        float s = 0.0f;
        for (int wv = 0; wv < 8; ++wv) s += smw[wv];
        partials[blockIdx.x] = s;
    }
}

// ---------------- kernel 4: final fixed-order reduction ----------------
__global__ void mmd_finalize(const float* __restrict__ partials,
                             float* __restrict__ out) {
    __shared__ float sm[256];
    int tid = threadIdx.x;
    float s = 0.0f;
    for (int k = 0; k < 16; ++k) s += partials[tid * 16 + k];
    sm[tid] = s;
    __syncthreads();
    for (int off = 128; off; off >>= 1) {
        if (tid < off) sm[tid] += sm[tid + off];
        __syncthreads();
    }
    if (tid == 0) out[0] = sm[0];
}

// ---------------- launcher ----------------
extern "C" void kernel_launch(void* const* d_in, const int* in_sizes, int n_in,
                              void* d_out, int out_size, void* d_ws, size_t ws_size,
                              hipStream_t stream) {
    const float* src = (const float*)d_in[0];
    const float* tgt = (const float*)d_in[1];
    float* out = (float*)d_out;

    float* wsf      = (float*)d_ws;
    float* norms    = wsf;             // 8192 floats
    float* bw       = wsf + 8192;      // 1 float
    float* partials = wsf + 8448;      // 4096 floats (total ws use ~49 KB)

    mmd_row_norms<<<1024, 256, 0, stream>>>(src, tgt, norms);
    mmd_bandwidth<<<1,    256, 0, stream>>>(src, tgt, norms, bw);
    mmd_main     <<<4096, 256, 0, stream>>>(src, tgt, norms, bw, partials);
    mmd_finalize <<<1,    256, 0, stream>>>(partials, out);
}